// GCN_12292196401223
// MI455X (gfx1250) — compile-verified
//
#include <hip/hip_runtime.h>

typedef __attribute__((ext_vector_type(2))) float v2f;
typedef __attribute__((ext_vector_type(8))) float v8f;

#define N_NODES  100000
#define N_EDGES  1600000
#define N_GRAPHS 512
#define IN_DIM   226
#define HID      128

// ---------------------------------------------------------------- utilities
__global__ void fill_kernel(float* __restrict__ p, float v, int n) {
    int i = blockIdx.x * blockDim.x + threadIdx.x;
    if (i < n) p[i] = v;
}

// deg[d] += 1 per edge (deg pre-initialized to 1.0 for the self loop)
__global__ void deg_kernel(const int* __restrict__ dst, float* __restrict__ deg, int E) {
    int e = blockIdx.x * blockDim.x + threadIdx.x;
    if (e < E) atomicAdd(&deg[dst[e]], 1.0f);
}

__global__ void rsqrt_kernel(float* __restrict__ d, int n) {
    int i = blockIdx.x * blockDim.x + threadIdx.x;
    if (i < n) d[i] = rsqrtf(d[i]);
}

// ---------------------------------------------------------------- B repack
// Bp[k4][t][lane] = { W[ka][n], W[ka+1][n] },  ka = 4*k4 + 2*(lane>=16),
// n = 16*t + (lane&15). Zero-padded for ka >= K, so the GEMM tail needs no
// B masking. Bp element = float2, fully coalesced b64 loads in the GEMM.
__global__ void pack_b_kernel(const float* __restrict__ W, float2* __restrict__ Bp, int K) {
    int i = blockIdx.x * blockDim.x + threadIdx.x;
    int K4 = (K + 3) >> 2;
    if (i >= K4 * 256) return;           // K4 * 8 tiles * 32 lanes
    int lane = i & 31;
    int t    = (i >> 5) & 7;
    int k4   = i >> 8;
    int ka   = k4 * 4 + ((lane >> 4) << 1);
    int n    = t * 16 + (lane & 15);
    float bx = (ka     < K) ? W[(size_t)ka * HID + n]       : 0.0f;
    float by = (ka + 1 < K) ? W[(size_t)(ka + 1) * HID + n] : 0.0f;
    Bp[i] = make_float2(bx, by);
}

// ---------------------------------------------------------------- WMMA GEMM
// H[n_rows x HID] = X[n_rows x K] @ W[K x HID], W pre-packed as Bp.
// One wave computes a 16-row x 128-col stripe with V_WMMA_F32_16X16X4_F32.
template<int K>
__global__ __launch_bounds__(256)
void gemm_wmma_kernel(const float* __restrict__ X, const v2f* __restrict__ Bp,
                      float* __restrict__ H, int n_rows) {
    constexpr int KMAIN = K & ~3;
    static_assert(KMAIN == K || (K - KMAIN) >= 2, "tail v2f load must stay in-row");

    const int lane = threadIdx.x & 31;
    const int wave = threadIdx.x >> 5;
    const int tile = blockIdx.x * 8 + wave;
    const int m0   = tile * 16;
    if (m0 >= n_rows) return;                 // wave-uniform exit: EXEC stays all-1s

    const int half = lane >> 4;               // 0: lanes 0-15, 1: lanes 16-31
    const int l15  = lane & 15;
    const int koff = half * 2;                // A-fragment K sub-offset per lane half

    const float* __restrict__ xrow = X + (size_t)(m0 + l15) * (size_t)K + koff;
    const v2f*   __restrict__ bp   = Bp + lane;

    v8f acc[8] = {};

    int k4 = 0;
    for (int k = 0; k < KMAIN; k += 4, ++k4) {
        v2f a = *(const v2f*)(xrow + k);                 // contiguous, 8B aligned
#pragma unroll
        for (int t = 0; t < 8; ++t) {
            v2f b = bp[k4 * 256 + t * 32];               // coalesced b64
            acc[t] = __builtin_amdgcn_wmma_f32_16x16x4_f32(
                false, a, false, b, (short)0, acc[t], false, false);
        }
    }

    if constexpr (KMAIN < K) {
        // Tail (K=226): lanes 16-31 would need K=226,227 -> select 0 branch-free.
        // All lanes load the in-range pair at column KMAIN (KMAIN+1 < K).
        const float* xt = X + (size_t)(m0 + l15) * (size_t)K + KMAIN;
        v2f at = *(const v2f*)xt;
        v2f a;
        a.x = half ? 0.0f : at.x;
        a.y = half ? 0.0f : at.y;
#pragma unroll
        for (int t = 0; t < 8; ++t) {
            v2f b = bp[k4 * 256 + t * 32];               // zero-padded rows
            acc[t] = __builtin_amdgcn_wmma_f32_16x16x4_f32(
                false, a, false, b, (short)0, acc[t], false, false);
        }
    }

#pragma unroll
    for (int t = 0; t < 8; ++t) {
#pragma unroll
        for (int r = 0; r < 8; ++r) {
            // C/D layout: VGPR r -> M = m0 + half*8 + r, N = l15
            H[(size_t)(m0 + half * 8 + r) * HID + t * 16 + l15] = acc[t][r];
        }
    }
}

// ---------------------------------------------------------------- graph ops
// A[i][:] = H[i][:] * dinv[i]^2   (self-loop term; also initializes accumulator)
__global__ __launch_bounds__(256)
void selfloop_kernel(const float* __restrict__ H, const float* __restrict__ dinv,
                     float* __restrict__ A, int n) {
    int idx = blockIdx.x * blockDim.x + threadIdx.x;   // n*32 threads, 4 ch each
    int i = idx >> 5;
    if (i >= n) return;
    int c = (idx & 31) * 4;
    float di = dinv[i];
    float w  = di * di;
    float4 v = *(const float4*)(H + (size_t)i * HID + c);
    *(float4*)(A + (size_t)i * HID + c) =
        make_float4(v.x * w, v.y * w, v.z * w, v.w * w);
}

// A[dst][:] += H[src][:] * dinv[src]*dinv[dst]   per edge (atomic f32 adds)
__global__ __launch_bounds__(256)
void scatter_kernel(const float* __restrict__ H, const int* __restrict__ src,
                    const int* __restrict__ dst, const float* __restrict__ dinv,
                    float* __restrict__ A, int E) {
    int idx = blockIdx.x * blockDim.x + threadIdx.x;   // E*32 threads
    int e = idx >> 5;
    if (e >= E) return;
    int c = (idx & 31) * 4;
    int s = src[e];
    int d = dst[e];
    float w = dinv[s] * dinv[d];
    float4 v = *(const float4*)(H + (size_t)s * HID + c);
    float* ap = A + (size_t)d * HID + c;
    atomicAdd(ap + 0, v.x * w);
    atomicAdd(ap + 1, v.y * w);
    atomicAdd(ap + 2, v.z * w);
    atomicAdd(ap + 3, v.w * w);
}

// A = relu(A + b)  in place
__global__ __launch_bounds__(256)
void bias_relu_kernel(float* __restrict__ A, const float* __restrict__ b, int n) {
    int idx = blockIdx.x * blockDim.x + threadIdx.x;
    int i = idx >> 5;
    if (i >= n) return;
    int c = (idx & 31) * 4;
    float4* p = (float4*)(A + (size_t)i * HID + c);
    float4 v = *p;
    float4 bb = *(const float4*)(b + c);
    v.x = fmaxf(v.x + bb.x, 0.0f);
    v.y = fmaxf(v.y + bb.y, 0.0f);
    v.z = fmaxf(v.z + bb.z, 0.0f);
    v.w = fmaxf(v.w + bb.w, 0.0f);
    *p = v;
}

// per-graph sums + counts
__global__ __launch_bounds__(256)
void pool_kernel(const float* __restrict__ X, const int* __restrict__ batch,
                 float* __restrict__ sums, float* __restrict__ cnt, int n) {
    int idx = blockIdx.x * blockDim.x + threadIdx.x;
    int i = idx >> 5;
    if (i >= n) return;
    int c = (idx & 31) * 4;
    int g = batch[i];
    float4 v = *(const float4*)(X + (size_t)i * HID + c);
    float* sp = sums + (size_t)g * HID + c;
    atomicAdd(sp + 0, v.x);
    atomicAdd(sp + 1, v.y);
    atomicAdd(sp + 2, v.z);
    atomicAdd(sp + 3, v.w);
    if (c == 0) atomicAdd(&cnt[g], 1.0f);
}

// out[g] = (sums[g]/max(cnt,1)) . Wl + bl
__global__ void final_kernel(const float* __restrict__ sums, const float* __restrict__ cnt,
                             const float* __restrict__ Wl, const float* __restrict__ bl,
                             float* __restrict__ out) {
    int g = blockIdx.x * blockDim.x + threadIdx.x;
    if (g >= N_GRAPHS) return;
    float c = fmaxf(cnt[g], 1.0f);
    float acc = 0.0f;
    for (int k = 0; k < HID; ++k) acc += sums[(size_t)g * HID + k] * Wl[k];
    out[g] = acc / c + bl[0];
}

// ---------------------------------------------------------------- launcher
extern "C" void kernel_launch(void* const* d_in, const int* in_sizes, int n_in,
                              void* d_out, int out_size, void* d_ws, size_t ws_size,
                              hipStream_t stream) {
    const float* x     = (const float*)d_in[0];
    const int*   ei    = (const int*)d_in[1];
    const int*   src   = ei;
    const int*   dst   = ei + N_EDGES;
    const int*   batch = (const int*)d_in[2];
    const float* W1 = (const float*)d_in[3];
    const float* b1 = (const float*)d_in[4];
    const float* W2 = (const float*)d_in[5];
    const float* b2 = (const float*)d_in[6];
    const float* W3 = (const float*)d_in[7];
    const float* b3 = (const float*)d_in[8];
    const float* Wl = (const float*)d_in[9];
    const float* bl = (const float*)d_in[10];
    float* out = (float*)d_out;

    // workspace layout
    float* buf0 = (float*)d_ws;                               // N*HID  (GEMM out H)
    float* buf1 = buf0 + (size_t)N_NODES * HID;               // N*HID  (scatter accum / activations)
    float* dinv = buf1 + (size_t)N_NODES * HID;               // N
    float* sums = dinv + N_NODES;                             // G*HID
    float* cnt  = sums + (size_t)N_GRAPHS * HID;              // G
    float* Bp   = cnt + N_GRAPHS;                             // <= 57*256*2 floats packed W

    const int TPB = 256;
    const int elem_threads = N_NODES * 32;                    // node x 4-ch chunks
    const int edge_threads = N_EDGES * 32;                    // edge x 4-ch chunks
    const int elem_blocks  = (elem_threads + TPB - 1) / TPB;
    const int edge_blocks  = (edge_threads + TPB - 1) / TPB;
    const int tiles        = (N_NODES + 15) / 16;
    const int gemm_blocks  = (tiles + 7) / 8;                 // 8 waves / block
    const int packK1       = ((IN_DIM + 3) / 4) * 256;        // threads for pack (K=226)
    const int packK2       = ((HID + 3) / 4) * 256;           // threads for pack (K=128)

    // degrees -> dinv  (init 1.0 accounts for the self loop)
    fill_kernel<<<(N_NODES + TPB - 1) / TPB, TPB, 0, stream>>>(dinv, 1.0f, N_NODES);
    fill_kernel<<<(N_GRAPHS * HID + TPB - 1) / TPB, TPB, 0, stream>>>(sums, 0.0f, N_GRAPHS * HID);
    fill_kernel<<<(N_GRAPHS + TPB - 1) / TPB, TPB, 0, stream>>>(cnt, 0.0f, N_GRAPHS);
    deg_kernel<<<(N_EDGES + TPB - 1) / TPB, TPB, 0, stream>>>(dst, dinv, N_EDGES);
    rsqrt_kernel<<<(N_NODES + TPB - 1) / TPB, TPB, 0, stream>>>(dinv, N_NODES);

    // ---- layer 1: x[N,226] -> buf1
    pack_b_kernel<<<(packK1 + TPB - 1) / TPB, TPB, 0, stream>>>(W1, (float2*)Bp, IN_DIM);
    gemm_wmma_kernel<IN_DIM><<<gemm_blocks, TPB, 0, stream>>>(x, (const v2f*)Bp, buf0, N_NODES);
    selfloop_kernel<<<elem_blocks, TPB, 0, stream>>>(buf0, dinv, buf1, N_NODES);
    scatter_kernel<<<edge_blocks, TPB, 0, stream>>>(buf0, src, dst, dinv, buf1, N_EDGES);
    bias_relu_kernel<<<elem_blocks, TPB, 0, stream>>>(buf1, b1, N_NODES);

    // ---- layer 2: buf1 -> buf1 (buf1 input consumed by GEMM before overwrite)
    pack_b_kernel<<<(packK2 + TPB - 1) / TPB, TPB, 0, stream>>>(W2, (float2*)Bp, HID);
    gemm_wmma_kernel<HID><<<gemm_blocks, TPB, 0, stream>>>(buf1, (const v2f*)Bp, buf0, N_NODES);
    selfloop_kernel<<<elem_blocks, TPB, 0, stream>>>(buf0, dinv, buf1, N_NODES);
    scatter_kernel<<<edge_blocks, TPB, 0, stream>>>(buf0, src, dst, dinv, buf1, N_EDGES);
    bias_relu_kernel<<<elem_blocks, TPB, 0, stream>>>(buf1, b2, N_NODES);

    // ---- layer 3
    pack_b_kernel<<<(packK2 + TPB - 1) / TPB, TPB, 0, stream>>>(W3, (float2*)Bp, HID);
    gemm_wmma_kernel<HID><<<gemm_blocks, TPB, 0, stream>>>(buf1, (const v2f*)Bp, buf0, N_NODES);
    selfloop_kernel<<<elem_blocks, TPB, 0, stream>>>(buf0, dinv, buf1, N_NODES);
    scatter_kernel<<<edge_blocks, TPB, 0, stream>>>(buf0, src, dst, dinv, buf1, N_EDGES);
    bias_relu_kernel<<<elem_blocks, TPB, 0, stream>>>(buf1, b3, N_NODES);

    // ---- global mean pool + final linear
    pool_kernel<<<elem_blocks, TPB, 0, stream>>>(buf1, batch, sums, cnt, N_NODES);
    final_kernel<<<(N_GRAPHS + 63) / 64, 64, 0, stream>>>(sums, cnt, Wl, bl, out);
}